// SAGE_46505905881800
// MI455X (gfx1250) — compile-verified
//
#include <hip/hip_runtime.h>
#include <hip/hip_bf16.h>

// ---------------------------------------------------------------------------
// 2-layer GraphSAGE for MI455X (gfx1250, wave32).
//   layer: mean-aggregate neighbors -> fused dual GEMM (mean@Wl.T + x@Wr.T + b)
//          via V_WMMA_F32_16X16X4_F32 -> L2 normalize (+ReLU after layer 1).
// Aggregation is the bandwidth bottleneck (L2-resident, 192MB L2 >> 51MB x);
// GEMM is only ~26 GFLOP total so we keep full f32 precision on the WMMA path.
// Two independent accumulator chains (c0: mean stream, c1: x stream) keep two
// WMMAs in flight in the XDL pipe instead of one serial RAW chain.
// ---------------------------------------------------------------------------

typedef __attribute__((ext_vector_type(2))) float v2f;
typedef __attribute__((ext_vector_type(8))) float v8f;

#define NNODES 50000
#define NEDGES 800000
#define DF     256      // D_IN == HID == 256
#define EPSV   1e-12f
#define ROWTILES (NNODES / 16)   // 3125, exact

// --------------------------- zero fill -------------------------------------
__global__ void sage_zero_kernel(float* __restrict__ p, long n) {
    long i = (long)blockIdx.x * blockDim.x + threadIdx.x;
    long stride = (long)gridDim.x * blockDim.x;
    for (; i < n; i += stride) p[i] = 0.0f;
}

// --------------------- edge scatter (mean numerator + degree) --------------
// One wave per edge: 32 lanes x 8 floats = 256-wide row. float4 gathers from
// the source row (L2-resident), 8 global_atomic_add_f32 per lane to agg[dst].
__global__ __launch_bounds__(256) void sage_scatter_kernel(
    const float* __restrict__ feat,
    const int*   __restrict__ src,
    const int*   __restrict__ dst,
    float* __restrict__ agg,
    float* cnt,              // may be nullptr (degree already counted)
    int nEdges) {
    int edge = blockIdx.x * 8 + (threadIdx.x >> 5);
    if (edge >= nEdges) return;          // wave-uniform
    int lane = threadIdx.x & 31;
    int s = src[edge];
    int d = dst[edge];

    const float4* fp = (const float4*)(feat + (size_t)s * DF + lane * 8);
    float4 a = fp[0];
    float4 b = fp[1];

    float* op = agg + (size_t)d * DF + lane * 8;
    atomicAdd(op + 0, a.x);
    atomicAdd(op + 1, a.y);
    atomicAdd(op + 2, a.z);
    atomicAdd(op + 3, a.w);
    atomicAdd(op + 4, b.x);
    atomicAdd(op + 5, b.y);
    atomicAdd(op + 6, b.z);
    atomicAdd(op + 7, b.w);

    if (cnt != nullptr && lane == 0) atomicAdd(cnt + d, 1.0f);
}

// ------------------------- agg / max(cnt,1) in place -----------------------
__global__ void sage_mean_kernel(float* __restrict__ agg,
                                 const float* __restrict__ cnt, long total) {
    long i = (long)blockIdx.x * blockDim.x + threadIdx.x;
    long stride = (long)gridDim.x * blockDim.x;
    for (; i < total; i += stride) {
        long row = i >> 8;  // DF == 256
        agg[i] = agg[i] / fmaxf(cnt[row], 1.0f);
    }
}

// --------------- fused dual GEMM:  out = mean@Wl.T + b + x@Wr.T ------------
// Block = 256 threads (8 waves). blockIdx.y selects a 16-col tile o0..o0+15;
// Wl/Wr column tiles are staged in LDS k-major [256][16] (conflict-free).
// Each wave computes one 16x16 output tile over K=256 with 128 f32 WMMAs
// split across two independent accumulator chains.
__global__ __launch_bounds__(256) void sage_gemm_kernel(
    const float* __restrict__ mean,
    const float* __restrict__ xin,
    const float* __restrict__ Wl,   // [256,256] row-major: Wl[o*256+k]
    const float* __restrict__ bl,   // [256]
    const float* __restrict__ Wr,
    float* __restrict__ out) {
    __shared__ float ldsB[2][DF * 16];   // [mat][k*16 + o], 32 KB total

    const int o0 = blockIdx.y * 16;

    // Cooperative stage: thread t loads Wl/Wr[(o0 + t%16), (t/16)*16 .. +15]
    // as 4x float4 and scatters transposed into LDS (k-major).
    {
        int t = threadIdx.x;
        int o = t & 15;
        int kb = (t >> 4) << 4;     // 0..240 step 16
        const float4* gl = (const float4*)(Wl + (size_t)(o0 + o) * DF + kb);
        const float4* gr = (const float4*)(Wr + (size_t)(o0 + o) * DF + kb);
#pragma unroll
        for (int j = 0; j < 4; ++j) {
            float4 vl = gl[j];
            float4 vr = gr[j];
            int k = kb + j * 4;
            ldsB[0][(k + 0) * 16 + o] = vl.x;
            ldsB[0][(k + 1) * 16 + o] = vl.y;
            ldsB[0][(k + 2) * 16 + o] = vl.z;
            ldsB[0][(k + 3) * 16 + o] = vl.w;
            ldsB[1][(k + 0) * 16 + o] = vr.x;
            ldsB[1][(k + 1) * 16 + o] = vr.y;
            ldsB[1][(k + 2) * 16 + o] = vr.z;
            ldsB[1][(k + 3) * 16 + o] = vr.w;
        }
    }
    __syncthreads();

    int rowTile = blockIdx.x * 8 + (threadIdx.x >> 5);
    if (rowTile >= ROWTILES) return;     // wave-uniform; no later barriers

    const int lane = threadIdx.x & 31;
    const int half = lane >> 4;          // 0: K={k,k+1}   1: K={k+2,k+3}
    const int l15  = lane & 15;

    const float* am = mean + ((size_t)rowTile * 16 + l15) * DF;
    const float* ax = xin  + ((size_t)rowTile * 16 + l15) * DF;

    // C/D layout: VGPR g -> row (g + 8*half), col l15. Bias depends on col only.
    float bv = bl[o0 + l15];
    v8f c0, c1;
#pragma unroll
    for (int g = 0; g < 8; ++g) { c0[g] = bv; c1[g] = 0.0f; }

#pragma unroll 4
    for (int k = 0; k < DF; k += 4) {
        int ka = k + half * 2;
        v2f A0; A0.x = am[ka]; A0.y = am[ka + 1];          // global_load_b64
        v2f A1; A1.x = ax[ka]; A1.y = ax[ka + 1];
        v2f B0; B0.x = ldsB[0][ka * 16 + l15]; B0.y = ldsB[0][(ka + 1) * 16 + l15];
        v2f B1; B1.x = ldsB[1][ka * 16 + l15]; B1.y = ldsB[1][(ka + 1) * 16 + l15];
        // 8 args: (neg_a, A, neg_b, B, c_mod, C, reuse_a, reuse_b)
        c0 = __builtin_amdgcn_wmma_f32_16x16x4_f32(false, A0, false, B0,
                                                   (short)0, c0, false, false);
        c1 = __builtin_amdgcn_wmma_f32_16x16x4_f32(false, A1, false, B1,
                                                   (short)0, c1, false, false);
    }

    float* orow = out + ((size_t)rowTile * 16 + half * 8) * DF + o0 + l15;
#pragma unroll
    for (int g = 0; g < 8; ++g) orow[(size_t)g * DF] = c0[g] + c1[g];
}

// ---------------------- row L2-normalize (+optional ReLU) ------------------
__global__ __launch_bounds__(256) void sage_norm_kernel(float* __restrict__ data,
                                                        int nRows, int doRelu) {
    int row = blockIdx.x * 8 + (threadIdx.x >> 5);
    if (row >= nRows) return;
    int lane = threadIdx.x & 31;
    float* p = data + (size_t)row * DF + lane * 8;
    float4 a = ((float4*)p)[0];
    float4 b = ((float4*)p)[1];
    float ss = a.x * a.x + a.y * a.y + a.z * a.z + a.w * a.w +
               b.x * b.x + b.y * b.y + b.z * b.z + b.w * b.w;
#pragma unroll
    for (int off = 16; off > 0; off >>= 1) ss += __shfl_xor(ss, off, 32);
    float scale = 1.0f / fmaxf(sqrtf(ss), EPSV);
    a.x *= scale; a.y *= scale; a.z *= scale; a.w *= scale;
    b.x *= scale; b.y *= scale; b.z *= scale; b.w *= scale;
    if (doRelu) {
        a.x = fmaxf(a.x, 0.f); a.y = fmaxf(a.y, 0.f);
        a.z = fmaxf(a.z, 0.f); a.w = fmaxf(a.w, 0.f);
        b.x = fmaxf(b.x, 0.f); b.y = fmaxf(b.y, 0.f);
        b.z = fmaxf(b.z, 0.f); b.w = fmaxf(b.w, 0.f);
    }
    ((float4*)p)[0] = a;
    ((float4*)p)[1] = b;
}

// ---------------------------------------------------------------------------
extern "C" void kernel_launch(void* const* d_in, const int* in_sizes, int n_in,
                              void* d_out, int out_size, void* d_ws, size_t ws_size,
                              hipStream_t stream) {
    const float* x    = (const float*)d_in[0];
    const int*   eidx = (const int*)d_in[1];      // [2, E]; src = row 0, dst = row 1
    const float* W1l  = (const float*)d_in[2];
    const float* b1l  = (const float*)d_in[3];
    const float* W1r  = (const float*)d_in[4];
    const float* W2l  = (const float*)d_in[5];
    const float* b2l  = (const float*)d_in[6];
    const float* W2r  = (const float*)d_in[7];
    float* out = (float*)d_out;

    const int* src = eidx;
    const int* dst = eidx + NEDGES;

    // Workspace layout (~103 MB): agg/mean [N*256] | cnt [N] | h [N*256]
    float* agg = (float*)d_ws;
    float* cnt = agg + (size_t)NNODES * DF;
    float* h   = cnt + NNODES;

    const long aggLen = (long)NNODES * DF;
    dim3 blk(256);
    dim3 gridScatter((NEDGES + 7) / 8);
    dim3 gridGemm((ROWTILES + 7) / 8, DF / 16);   // (391, 16)
    dim3 gridNorm((NNODES + 7) / 8);

    // ----- layer 1 -----
    sage_zero_kernel<<<2048, blk, 0, stream>>>(agg, aggLen + NNODES);  // agg + cnt
    sage_scatter_kernel<<<gridScatter, blk, 0, stream>>>(x, src, dst, agg, cnt, NEDGES);
    sage_mean_kernel<<<4096, blk, 0, stream>>>(agg, cnt, aggLen);
    sage_gemm_kernel<<<gridGemm, blk, 0, stream>>>(agg, x, W1l, b1l, W1r, h);
    sage_norm_kernel<<<gridNorm, blk, 0, stream>>>(h, NNODES, /*doRelu=*/1);

    // ----- layer 2 (degrees unchanged; reuse cnt) -----
    sage_zero_kernel<<<2048, blk, 0, stream>>>(agg, aggLen);
    sage_scatter_kernel<<<gridScatter, blk, 0, stream>>>(h, src, dst, agg, nullptr, NEDGES);
    sage_mean_kernel<<<4096, blk, 0, stream>>>(agg, cnt, aggLen);
    sage_gemm_kernel<<<gridGemm, blk, 0, stream>>>(agg, h, W2l, b2l, W2r, out);
    sage_norm_kernel<<<gridNorm, blk, 0, stream>>>(out, NNODES, /*doRelu=*/0);
}